// GCNEncoder_89644557403153
// MI455X (gfx1250) — compile-verified
//
#include <hip/hip_runtime.h>
#include <hip/hip_bf16.h>

typedef __attribute__((ext_vector_type(2))) float v2f;
typedef __attribute__((ext_vector_type(8))) float v8f;

#define NODE_C 256   // HID_C == OUT_C == 256

// ---------------------------------------------------------------------------
// Degree / normalization kernels
// ---------------------------------------------------------------------------
__global__ void k_fill1(float* __restrict__ deg, int n) {
    int i = blockIdx.x * blockDim.x + threadIdx.x;
    if (i < n) deg[i] = 1.0f;   // self-loop contribution
}

__global__ void k_deg(const long long* __restrict__ dst, float* __restrict__ deg,
                      long long nE) {
    long long e = (long long)blockIdx.x * blockDim.x + threadIdx.x;
    if (e < nE) {
#if defined(__HIP_DEVICE_COMPILE__)
        unsafeAtomicAdd(&deg[(int)dst[e]], 1.0f);
#endif
    }
}

__global__ void k_invsqrt(float* __restrict__ dis, int n) {
    int i = blockIdx.x * blockDim.x + threadIdx.x;
    if (i < n) {
        float d = dis[i];
        dis[i] = (d > 0.0f) ? (1.0f / sqrtf(d)) : 0.0f;
    }
}

// ---------------------------------------------------------------------------
// GEMM: out[N, 256] = (RELU ? relu(A) : A)[N, CIN] @ W[CIN, 256]
// grid = (N/16, 2), block = 256 (8 waves). Wave w computes the 16x16 tile at
// columns (blockIdx.y*128 + w*16). A tile staged in LDS with padded stride.
// Uses V_WMMA_F32_16X16X4_F32, chaining K in steps of 4.
// ---------------------------------------------------------------------------
template <int CIN, bool RELU>
__global__ __launch_bounds__(256)
void k_gemm(const float* __restrict__ A, const float* __restrict__ W,
            float* __restrict__ out, int n_rows) {
    constexpr int LDSROW = CIN + 2;           // pad: conflict-free + float2 aligned
    __shared__ float sA[16 * LDSROW];

    const int m0  = blockIdx.x * 16;
    const int tid = threadIdx.x;

    // ---- cooperative stage of A tile (coalesced along rows) ----
    #pragma unroll
    for (int e = tid; e < 16 * CIN; e += 256) {
        int r = e / CIN, c = e % CIN;
        float v = A[(size_t)(m0 + r) * CIN + c];
        if (RELU) v = fmaxf(v, 0.0f);
        sA[r * LDSROW + c] = v;
    }
    __syncthreads();

    const int wave = tid >> 5;
    const int lane = tid & 31;
    const int half = lane >> 4;               // 0: K pair {k,k+1}; 1: {k+2,k+3}
    const int arow = lane & 15;               // A-matrix row M within tile
    const int col  = blockIdx.y * 128 + wave * 16 + (lane & 15); // B/D column

    const float* sArow = &sA[arow * LDSROW + half * 2];

    v8f acc = {};
    #pragma unroll 8
    for (int k = 0; k < CIN; k += 4) {
        v2f a;
        a.x = sArow[k];
        a.y = sArow[k + 1];
        v2f b;
        b.x = W[(size_t)(k + half * 2)     * NODE_C + col];
        b.y = W[(size_t)(k + half * 2 + 1) * NODE_C + col];
        acc = __builtin_amdgcn_wmma_f32_16x16x4_f32(
            false, a, false, b, (short)0, acc, false, false);
    }

    // ---- store D: VGPR r -> row m0 + r + half*8, column col ----
    #pragma unroll
    for (int r = 0; r < 8; ++r) {
        out[(size_t)(m0 + r + half * 8) * NODE_C + col] = acc[r];
    }
}

// ---------------------------------------------------------------------------
// Aggregation init: out[i][c] = b[c] + dis[i]^2 * xw[i][c]   (self-loop + bias)
// One thread per float4 chunk: n*64 threads.
// ---------------------------------------------------------------------------
__global__ __launch_bounds__(256)
void k_agg_init(const float* __restrict__ xw, const float* __restrict__ dis,
                const float* __restrict__ bias, float* __restrict__ out, int n) {
    long long t = (long long)blockIdx.x * blockDim.x + threadIdx.x;
    if (t >= (long long)n * 64) return;
    int node = (int)(t >> 6);
    int c4   = (int)(t & 63) * 4;
    float d    = dis[node];
    float coef = d * d;
    const float4 v  = *(const float4*)&xw[(size_t)node * NODE_C + c4];
    const float4 bb = *(const float4*)&bias[c4];
    float4 o;
    o.x = bb.x + coef * v.x;
    o.y = bb.y + coef * v.y;
    o.z = bb.z + coef * v.z;
    o.w = bb.w + coef * v.w;
    *(float4*)&out[(size_t)node * NODE_C + c4] = o;
}

// ---------------------------------------------------------------------------
// Edge aggregation: out[dst] += dis[src]*dis[dst] * xw[src]
// 64 threads per edge, float4 per thread, fp32 hardware atomics.
// ---------------------------------------------------------------------------
__global__ __launch_bounds__(256)
void k_agg_edges(const float* __restrict__ xw, const float* __restrict__ dis,
                 const long long* __restrict__ src, const long long* __restrict__ dst,
                 float* __restrict__ out, long long nE) {
    long long t = (long long)blockIdx.x * blockDim.x + threadIdx.x;
    long long e = t >> 6;
    if (e >= nE) return;
    int c4 = (int)(t & 63) * 4;
    int s  = (int)src[e];
    int d  = (int)dst[e];
    float coef = dis[s] * dis[d];
    const float4 v = *(const float4*)&xw[(size_t)s * NODE_C + c4];
    float* o = &out[(size_t)d * NODE_C + c4];
#if defined(__HIP_DEVICE_COMPILE__)
    unsafeAtomicAdd(o + 0, coef * v.x);
    unsafeAtomicAdd(o + 1, coef * v.y);
    unsafeAtomicAdd(o + 2, coef * v.z);
    unsafeAtomicAdd(o + 3, coef * v.w);
#endif
}

// ---------------------------------------------------------------------------
// Driver
// ---------------------------------------------------------------------------
extern "C" void kernel_launch(void* const* d_in, const int* in_sizes, int n_in,
                              void* d_out, int out_size, void* d_ws, size_t ws_size,
                              hipStream_t stream) {
    const float*     x  = (const float*)d_in[0];
    const long long* ei = (const long long*)d_in[1];
    const float*     W0 = (const float*)d_in[2];
    const float*     b0 = (const float*)d_in[3];
    const float*     W1 = (const float*)d_in[4];
    const float*     b1 = (const float*)d_in[5];
    const float*     W2 = (const float*)d_in[6];
    const float*     b2 = (const float*)d_in[7];
    float*           out = (float*)d_out;

    const int       N = in_sizes[0] / 128;   // 100000 (multiple of 16)
    const long long E = in_sizes[1] / 2;     // 1600000
    const long long* src = ei;
    const long long* dst = ei + E;

    // workspace layout: dis [N] | xw [N*256] | h [N*256]
    char* ws = (char*)d_ws;
    size_t off = 0;
    float* dis = (float*)(ws + off);
    off += (((size_t)N * sizeof(float)) + 255) & ~(size_t)255;
    float* xw = (float*)(ws + off);
    off += (size_t)N * NODE_C * sizeof(float);
    float* h  = (float*)(ws + off);

    const int nthr = 256;
    const int nBlkN   = (N + nthr - 1) / nthr;
    const int nBlkE   = (int)((E + nthr - 1) / nthr);
    const int nBlkAgg = (int)(((long long)N * 64 + nthr - 1) / nthr);
    const int nBlkEdg = (int)((E * 64 + nthr - 1) / nthr);
    dim3 gemmGrid(N / 16, NODE_C / 128);

    // --- degree / normalization (once; shared by all 3 layers) ---
    k_fill1  <<<nBlkN, nthr, 0, stream>>>(dis, N);
    k_deg    <<<nBlkE, nthr, 0, stream>>>(dst, dis, E);
    k_invsqrt<<<nBlkN, nthr, 0, stream>>>(dis, N);

    // --- layer 0: x[128] -> h[256], relu deferred to next GEMM's read ---
    k_gemm<128, false><<<gemmGrid, nthr, 0, stream>>>(x, W0, xw, N);
    k_agg_init <<<nBlkAgg, nthr, 0, stream>>>(xw, dis, b0, h, N);
    k_agg_edges<<<nBlkEdg, nthr, 0, stream>>>(xw, dis, src, dst, h, E);

    // --- layer 1: relu(h)[256] -> h[256] ---
    k_gemm<256, true><<<gemmGrid, nthr, 0, stream>>>(h, W1, xw, N);
    k_agg_init <<<nBlkAgg, nthr, 0, stream>>>(xw, dis, b1, h, N);
    k_agg_edges<<<nBlkEdg, nthr, 0, stream>>>(xw, dis, src, dst, h, E);

    // --- layer 2: relu(h)[256] -> out[256], no relu on output ---
    k_gemm<256, true><<<gemmGrid, nthr, 0, stream>>>(h, W2, xw, N);
    k_agg_init <<<nBlkAgg, nthr, 0, stream>>>(xw, dis, b2, out, N);
    k_agg_edges<<<nBlkEdg, nthr, 0, stream>>>(xw, dis, src, dst, out, E);
}